// StyleGAN4AugmentPipe_60035052863529
// MI455X (gfx1250) — compile-verified
//
#include <hip/hip_runtime.h>
#include <hip/hip_bf16.h>
#include <math.h>

// Problem constants (match reference)
#define BATCH 32
#define CHN 3
#define IMH 512
#define IMW 512
#define PLANE (IMH * IMW)          // 262144
#define CUT 128                    // int(512 * 0.25)
#define BRIGHTNESS 0.2f
#define CONTRAST 0.2f
#define HUEAMP 0.1f
#define SATAMP 0.2f
#define ROTRAD 0.17453292519943295f // 10 deg in rad

#define TILE 32                    // output tile edge
#define CAP 44                     // LDS source-box capacity (>= 41 needed at 10deg)

typedef unsigned int v4u __attribute__((ext_vector_type(4)));
typedef int v8i __attribute__((ext_vector_type(8)));
typedef int v4i __attribute__((ext_vector_type(4)));

static __device__ __forceinline__ unsigned rfl_u(unsigned v) {
  return (unsigned)__builtin_amdgcn_readfirstlane((int)v);
}
static __device__ __forceinline__ int rfl_i(int v) {
  return __builtin_amdgcn_readfirstlane(v);
}

static __device__ __forceinline__ float clampf1(float v, float lo, float hi) {
  return fminf(fmaxf(v, lo), hi);
}
static __device__ __forceinline__ float mod360f(float a) {
  return a - floorf(a * (1.0f / 360.0f)) * 360.0f;
}

// In-place bright->contrast->HSV adjust for one RGB triplet.
static __device__ __forceinline__ void pixel_transform(
    float& r, float& g, float& b,
    float bf, float cf, float m0, float m1, float m2,
    float hshift, float sfac) {
  // brightness
  r = clampf1(r + bf, -1.f, 1.f);
  g = clampf1(g + bf, -1.f, 1.f);
  b = clampf1(b + bf, -1.f, 1.f);
  // contrast (per-channel plane mean)
  r = clampf1((r - m0) * cf + m0, -1.f, 1.f);
  g = clampf1((g - m1) * cf + m1, -1.f, 1.f);
  b = clampf1((b - m2) * cf + m2, -1.f, 1.f);
  // rgb -> hsv (exact reference formulas)
  float mx = fmaxf(r, fmaxf(g, b));
  float mn = fminf(r, fminf(g, b));
  float diff = mx - mn;
  float sd = (diff == 0.0f) ? 1.0f : diff;
  float hr = mod360f(60.0f * ((g - b) / sd)) * (1.0f / 360.0f);
  float hg = mod360f(60.0f * ((b - r) / sd + 2.0f)) * (1.0f / 360.0f);
  float hb = mod360f(60.0f * ((r - g) / sd + 4.0f)) * (1.0f / 360.0f);
  float h = (mx == b) ? hb : ((mx == g) ? hg : ((mx == r) ? hr : 0.0f));
  float s = (mx != 0.0f) ? diff / mx : 0.0f;
  float v = mx;
  // hue / sat adjust
  h = clampf1(h + hshift, 0.0f, 1.0f);
  s = clampf1(s * sfac, 0.0f, 1.0f);
  // hsv -> rgb
  float H = h * 360.0f;
  float c = v * s;
  float Hd = H * (1.0f / 60.0f);
  float X = c * (1.0f - fabsf((Hd - floorf(Hd * 0.5f) * 2.0f) - 1.0f));
  float m = v - c;
  float rr, gg, bb;
  if (H < 60.0f)       { rr = c; gg = X; bb = 0.f; }
  else if (H < 120.0f) { rr = X; gg = c; bb = 0.f; }
  else if (H < 180.0f) { rr = 0.f; gg = c; bb = X; }
  else if (H < 240.0f) { rr = 0.f; gg = X; bb = c; }
  else if (H < 300.0f) { rr = X; gg = 0.f; bb = c; }
  else if (H < 360.0f) { rr = c; gg = 0.f; bb = X; }
  else                 { rr = 0.f; gg = 0.f; bb = 0.f; } // H == 360 -> select default
  r = rr + m; g = gg + m; b = bb + m;
}

// Rotated pixel-space coordinates for output pixel (w,h).
static __device__ __forceinline__ float2 rot_pix(float ca, float sa, int w, int h) {
  float xn = (float)w * (2.0f / 511.0f) - 1.0f;
  float yn = (float)h * (2.0f / 511.0f) - 1.0f;
  float gx = ca * xn - sa * yn;
  float gy = sa * xn + ca * yn;
  return make_float2((gx + 1.0f) * 256.0f - 0.5f, (gy + 1.0f) * 256.0f - 0.5f);
}

// ---- Kernel 1: per-plane partial sums of clip(img + bf, -1, 1) ----
// mean is invariant under the vertical flip, so read linearly (coalesced).
__global__ void __launch_bounds__(256) mean_partial_kernel(
    const float* __restrict__ img, const float* __restrict__ u_bright,
    float* __restrict__ partials) {
  const int plane = blockIdx.x >> 3;  // 0..95  (b*3 + c)
  const int part = blockIdx.x & 7;    // 0..7
  const int b = plane / 3;
  const float bf = (u_bright[b] * 2.0f - 1.0f) * BRIGHTNESS;
  const float4* p4 =
      (const float4*)(img + (size_t)plane * PLANE + (size_t)part * 32768);
  const int t = threadIdx.x;
  float s = 0.0f;
  for (int i = t; i < 8192; i += 256) {  // 8192 float4 = 32768 floats
    float4 v = p4[i];
    s += clampf1(v.x + bf, -1.f, 1.f) + clampf1(v.y + bf, -1.f, 1.f) +
         clampf1(v.z + bf, -1.f, 1.f) + clampf1(v.w + bf, -1.f, 1.f);
  }
  __shared__ float red[256];
  red[t] = s;
  __syncthreads();
  for (int off = 128; off > 0; off >>= 1) {
    if (t < off) red[t] += red[t + off];
    __syncthreads();
  }
  if (t == 0) partials[blockIdx.x] = red[0];
}

// ---- Kernel 2: deterministic finalize of 96 plane means ----
__global__ void mean_final_kernel(const float* __restrict__ partials,
                                  float* __restrict__ means) {
  int i = threadIdx.x;
  if (i < BATCH * CHN) {
    float s = 0.0f;
    for (int p = 0; p < 8; ++p) s += partials[i * 8 + p];
    means[i] = s * (1.0f / (float)PLANE);
  }
}

// ---- Kernel 3: fused flip+bright+contrast+HSV+rotate+noise+cutout ----
// One block = one 32x32 output tile of one image. The rotated-source bounding
// box (clipped to the image) is DMA'd into LDS by the Tensor Data Mover
// (3 x TENSOR_LOAD_TO_LDS, one per channel), transformed in place once per
// source cell, then bilinearly gathered. Vertical flip is applied at gather
// time so the TDM copy is a plain strided 2D tile.
__global__ void __launch_bounds__(256) augment_kernel(
    const float* __restrict__ img, const float* __restrict__ u_b,
    const float* __restrict__ u_c, const float* __restrict__ u_h,
    const float* __restrict__ u_s, const float* __restrict__ u_a,
    const float* __restrict__ noise, const int* __restrict__ cxy,
    const unsigned char* __restrict__ capply, const float* __restrict__ means,
    float* __restrict__ out) {
  const int b = blockIdx.z;
  const int w0 = blockIdx.x * TILE;
  const int h0 = blockIdx.y * TILE;
  const int tid = threadIdx.x;

  // per-batch uniforms
  const float bf = (u_b[b] * 2.0f - 1.0f) * BRIGHTNESS;
  const float cf = 1.0f + (u_c[b] * 2.0f - 1.0f) * CONTRAST;
  const float hshift = (u_h[b] * 2.0f - 1.0f) * HUEAMP;
  const float sfac = 1.0f + (u_s[b] * 2.0f - 1.0f) * SATAMP;
  const float ang = (u_a[b] * 2.0f - 1.0f) * ROTRAD;
  const float ca = cosf(ang), sa = sinf(ang);
  const float m0 = means[b * 3 + 0];
  const float m1 = means[b * 3 + 1];
  const float m2 = means[b * 3 + 2];

  // source bounding box from tile corners (map is linear -> corners suffice)
  float2 c00 = rot_pix(ca, sa, w0, h0);
  float2 c10 = rot_pix(ca, sa, w0 + TILE - 1, h0);
  float2 c01 = rot_pix(ca, sa, w0, h0 + TILE - 1);
  float2 c11 = rot_pix(ca, sa, w0 + TILE - 1, h0 + TILE - 1);
  float mnx = fminf(fminf(c00.x, c10.x), fminf(c01.x, c11.x));
  float mxx = fmaxf(fmaxf(c00.x, c10.x), fmaxf(c01.x, c11.x));
  float mny = fminf(fminf(c00.y, c10.y), fminf(c01.y, c11.y));
  float mxy = fmaxf(fmaxf(c00.y, c10.y), fmaxf(c01.y, c11.y));
  const int bx0 = (int)floorf(mnx) - 1;
  const int by0 = (int)floorf(mny) - 1;
  int BW = (int)floorf(mxx) - bx0 + 3;  // floor(max)+1 neighbor + 1 margin
  int BH = (int)floorf(mxy) - by0 + 3;
  if (BW > CAP) BW = CAP;
  if (BH > CAP) BH = CAP;

  // Clip the box to the image. Clipped-away cells are never gathered
  // (the bilinear gather re-checks coordinate validity), so the TDM copy
  // needs no OOB handling and never reads outside the buffer.
  const int bxl = max(bx0, 0);
  const int bxr = min(bx0 + BW - 1, IMW - 1);
  const int rlo = max(IMH - 1 - (by0 + BH - 1), 0);  // memory rows ascending
  const int rhi = min(IMH - 1 - by0, IMH - 1);
  int BWt = bxr - bxl + 1;
  int BHt = rhi - rlo + 1;
  if (BWt < 0) BWt = 0;
  if (BHt < 0) BHt = 0;
  const int ncell = BWt * BHt;  // <= 41*41 < CAP*CAP

  __shared__ float tile[3 * CAP * CAP];
  const unsigned ldsbase = (unsigned)(size_t)(&tile[0]);  // LDS byte offset

  // Phase 1: Tensor Data Mover DMA, one 2D tile per channel, issued by the
  // first wave only (TDM ignores EXEC; branch is scalar-uniform).
  if (rfl_i((int)tid) == 0 && ncell > 0) {
    const unsigned tdw = rfl_u((unsigned)BWt);   // tile/tensor dim0 (elements)
    const unsigned tdh = rfl_u((unsigned)BHt);   // tile/tensor dim1 (rows)
#pragma unroll
    for (int c = 0; c < 3; ++c) {
      unsigned long long ga =
          (unsigned long long)(const void*)img +
          ((((unsigned long long)(b * 3 + c) * IMH + (unsigned long long)rlo) *
                IMW +
            (unsigned long long)bxl) *
           4ull);
      const unsigned lds_addr = rfl_u(ldsbase + (unsigned)(c * CAP * CAP * 4));
      const unsigned ga_lo = rfl_u((unsigned)(ga & 0xFFFFFFFFull));
      const unsigned ga_hi =
          rfl_u(((unsigned)(ga >> 32) & 0x01FFFFFFu) | 0x80000000u); // type=2
      // D# group 0: count=1 | lds_addr | global_addr[56:0] | type=2
      v4u g0 = {1u, lds_addr, ga_lo, ga_hi};
      // D# group 1: data_size=4B; tensor_dim0/1 == tile_dim0/1 (pre-clipped);
      // tensor_dim0_stride = 512 elements (image row pitch).
      v8i g1 = {
          (int)0x00020000u,                               // data_size=2 (4B)
          (int)((tdw & 0xFFFFu) << 16),                   // tensor_dim0 lo16
          (int)(((tdw >> 16) & 0xFFFFu) |                 // tensor_dim0 hi16
                ((tdh & 0xFFFFu) << 16)),                 // tensor_dim1 lo16
          (int)(((tdh >> 16) & 0xFFFFu) |                 // tensor_dim1 hi16
                ((tdw & 0xFFFFu) << 16)),                 // tile_dim0
          (int)(tdh & 0xFFFFu),                           // tile_dim1 (dim2=0)
          (int)(unsigned)IMW,                             // dim0 stride lo32
          0, 0};                                          // stride hi / dim1str
      v4i gz = {0, 0, 0, 0};                              // 2D: groups 2,3 = 0
      v8i gz8 = {0, 0, 0, 0, 0, 0, 0, 0};                 // unused ext group
      __builtin_amdgcn_tensor_load_to_lds(g0, g1, gz, gz, gz8, 0);
    }
    __builtin_amdgcn_s_wait_tensorcnt(0);
  }
  __syncthreads();

  // Phase 2: pointwise transform, in place, once per source cell
  for (int idx = tid; idx < ncell; idx += 256) {
    float r = tile[0 * CAP * CAP + idx];
    float g = tile[1 * CAP * CAP + idx];
    float bb = tile[2 * CAP * CAP + idx];
    pixel_transform(r, g, bb, bf, cf, m0, m1, m2, hshift, sfac);
    tile[0 * CAP * CAP + idx] = r;
    tile[1 * CAP * CAP + idx] = g;
    tile[2 * CAP * CAP + idx] = bb;
  }
  __syncthreads();

  // Phase 3: bilinear gather (with vertical flip) + noise + cutout + store
  const int cutx = cxy[2 * b + 0];
  const int cuty = cxy[2 * b + 1];
  const bool doCut = capply[b] != 0;
#pragma unroll
  for (int k = 0; k < 4; ++k) {
    int pix = tid + k * 256;
    int oy = h0 + (pix >> 5);
    int ox = w0 + (pix & 31);
    size_t obase = ((size_t)(b * 3) * IMH + (size_t)oy) * IMW + (size_t)ox;
    __builtin_prefetch(noise + obase, 0, 0);  // global_prefetch_b8
    float2 gp = rot_pix(ca, sa, ox, oy);
    float x0f = floorf(gp.x), y0f = floorf(gp.y);
    float wx = gp.x - x0f, wy = gp.y - y0f;
    int x0 = (int)x0f, y0 = (int)y0f;
    float a0 = 0.f, a1 = 0.f, a2 = 0.f;
#pragma unroll
    for (int dy = 0; dy < 2; ++dy) {
#pragma unroll
      for (int dx = 0; dx < 2; ++dx) {
        int xi = x0 + dx, yi = y0 + dy;
        float wgt = (dy ? wy : 1.0f - wy) * (dx ? wx : 1.0f - wx);
        if (xi >= 0 && xi < IMW && yi >= 0 && yi < IMH) {
          // flipped-source row yi lives at memory row (511 - yi)
          int lidx = ((IMH - 1 - yi) - rlo) * BWt + (xi - bxl);
          a0 += wgt * tile[0 * CAP * CAP + lidx];
          a1 += wgt * tile[1 * CAP * CAP + lidx];
          a2 += wgt * tile[2 * CAP * CAP + lidx];
        }
      }
    }
    bool cut = doCut && (oy >= cuty) && (oy < cuty + CUT) && (ox >= cutx) &&
               (ox < cutx + CUT);
    float acc[3] = {a0, a1, a2};
#pragma unroll
    for (int c = 0; c < 3; ++c) {
      size_t off = obase + (size_t)c * PLANE;
      float v = clampf1(acc[c] + noise[off] * 0.05f, -1.0f, 1.0f);
      out[off] = cut ? 0.0f : v;
    }
  }
}

extern "C" void kernel_launch(void* const* d_in, const int* in_sizes, int n_in,
                              void* d_out, int out_size, void* d_ws,
                              size_t ws_size, hipStream_t stream) {
  const float* img = (const float*)d_in[0];
  const float* u_bright = (const float*)d_in[1];
  const float* u_contrast = (const float*)d_in[2];
  const float* u_hue = (const float*)d_in[3];
  const float* u_sat = (const float*)d_in[4];
  const float* u_angle = (const float*)d_in[5];
  const float* noise = (const float*)d_in[6];
  const int* cutout_xy = (const int*)d_in[7];
  const unsigned char* cutout_apply = (const unsigned char*)d_in[8];

  float* ws = (float*)d_ws;
  float* partials = ws;        // 768 floats
  float* means = ws + 768;     // 96 floats

  mean_partial_kernel<<<BATCH * CHN * 8, 256, 0, stream>>>(img, u_bright,
                                                           partials);
  mean_final_kernel<<<1, 128, 0, stream>>>(partials, means);

  dim3 grid(IMW / TILE, IMH / TILE, BATCH);  // 16 x 16 x 32
  augment_kernel<<<grid, 256, 0, stream>>>(img, u_bright, u_contrast, u_hue,
                                           u_sat, u_angle, noise, cutout_xy,
                                           cutout_apply, means, (float*)d_out);
}